// GCN_81363860455527
// MI455X (gfx1250) — compile-verified
//
#include <hip/hip_runtime.h>

#define NNODES 100000
#define NEDGES 1600000
#define MTILES (NNODES / 16)   // 6250

typedef float v2f __attribute__((ext_vector_type(2)));
typedef float v8f __attribute__((ext_vector_type(8)));

__device__ __forceinline__ void atomAddF(float* p, float v) {
    __hip_atomic_fetch_add(p, v, __ATOMIC_RELAXED, __HIP_MEMORY_SCOPE_AGENT);
}
__device__ __forceinline__ void atomAddLDS(float* p, float v) {
    __hip_atomic_fetch_add(p, v, __ATOMIC_RELAXED, __HIP_MEMORY_SCOPE_WORKGROUP);
}

// ---------------------------------------------------------------------------
// deg[dst] += 1 for every edge; E/256 = 6250 blocks exactly.
__global__ __launch_bounds__(256) void degree_kernel(
    const long long* __restrict__ rowi, float* __restrict__ deg) {
    const unsigned e = blockIdx.x * 256u + threadIdx.x;
    const int dst = (int)rowi[e];
    atomAddF(&deg[dst], 1.0f);
}

// deg -> 1/(deg + eps), in place.
__global__ __launch_bounds__(256) void rdeg_kernel(float* __restrict__ deg, int n) {
    const int i = blockIdx.x * 256 + threadIdx.x;
    if (i < n) deg[i] = 1.0f / (deg[i] + 1e-6f);
}

// ---------------------------------------------------------------------------
// agg[row[e]][*] += x[col[e]][*]; one thread per (edge, 4 features).
// float4 gather (contiguous 128B+ per edge across lanes) + 4x global f32 atomics.
template<int D>
__global__ __launch_bounds__(256) void scatter_add_kernel(
    const long long* __restrict__ rowi, const long long* __restrict__ coli,
    const float* __restrict__ X, float* __restrict__ AGG) {
    constexpr unsigned CPE = D / 4;            // chunks per edge (16 or 32)
    const unsigned gid   = blockIdx.x * 256u + threadIdx.x;
    const unsigned e     = gid / CPE;          // power of two -> shift
    const unsigned chunk = gid & (CPE - 1u);
    const int dst = (int)rowi[e];
    const int src = (int)coli[e];
    const float4 v = *(const float4*)(X + src * D + chunk * 4);
    float* p = AGG + dst * D + chunk * 4;
    atomAddF(p + 0, v.x);
    atomAddF(p + 1, v.y);
    atomAddF(p + 2, v.z);
    atomAddF(p + 3, v.w);
}

// ---------------------------------------------------------------------------
// H[n,o] = sum_k (A[n,k]*rdeg[n]) * W[o,k] + bias[o], 16 rows per wave,
// ALL output tiles per wave (A-frag loaded once -> 8x less L2 A-traffic).
// V_WMMA_F32_16X16X4_F32; A/B frag: lane L holds row base+(L&15),
// cols k0+2*(L>>4)..+1 (one float2 load). C: VGPR r -> (M=r|r+8, N=lane&15).
// BN stats: shfl_xor(16) half-wave combine -> LDS ds_add_f32 -> one global
// atomic flush of 256 partials per block.
template<int K, int OUTD, bool STATS>
__global__ __launch_bounds__(256) void gemm_bias_stats(
    const float* __restrict__ A, const float* __restrict__ rdeg,
    const float* __restrict__ W, const float* __restrict__ bias,
    float* __restrict__ H, float* __restrict__ stats) {
    constexpr int NTO = OUTD / 16;
    __shared__ float sred[2 * OUTD];

    const int lane  = threadIdx.x & 31;
    const int l15   = lane & 15;
    const int hi    = lane >> 4;               // 0 or 1
    const int mtile = blockIdx.x * 8 + (threadIdx.x >> 5);
    const bool valid = (mtile < MTILES);
    const int mcl   = valid ? mtile : (MTILES - 1);   // clamp loads, keep EXEC=all-1
    const int row   = mcl * 16 + l15;
    const int koff  = hi * 2;
    const float rd  = rdeg[row];
    const float* ap = A + row * K + koff;
    const float* wp = W + l15 * K + koff;

    if (STATS) {
        if (threadIdx.x < 2 * OUTD) sred[threadIdx.x] = 0.f;
        __syncthreads();
    }

    v8f acc[NTO];
#pragma unroll
    for (int t = 0; t < NTO; ++t) acc[t] = v8f{0.f,0.f,0.f,0.f,0.f,0.f,0.f,0.f};

#pragma unroll
    for (int k0 = 0; k0 < K; k0 += 4) {
        v2f a = *(const v2f*)(ap + k0);
        a *= rd;                               // fold 1/deg into A rows
#pragma unroll
        for (int t = 0; t < NTO; ++t) {
            v2f b = *(const v2f*)(wp + t * 16 * K + k0);
            acc[t] = __builtin_amdgcn_wmma_f32_16x16x4_f32(
                false, a, false, b, (short)0, acc[t], false, false);
        }
    }

    const int rbase = mcl * 16 + hi * 8;
#pragma unroll
    for (int t = 0; t < NTO; ++t) {
        const int ocol = t * 16 + l15;
        const float bb = bias[ocol];
        float s = 0.f, s2 = 0.f;
#pragma unroll
        for (int r = 0; r < 8; ++r) {
            float v = acc[t][r] + bb;
            if (valid) H[(rbase + r) * OUTD + ocol] = v;
            s += v; s2 += v * v;
        }
        if (STATS) {
            s  += __shfl_xor(s, 16, 32);       // combine the two half-waves
            s2 += __shfl_xor(s2, 16, 32);
            if (valid && hi == 0) {            // lanes 0..15 own 16 columns
                atomAddLDS(&sred[ocol], s);
                atomAddLDS(&sred[OUTD + ocol], s2);
            }
        }
    }
    if (STATS) {
        __syncthreads();
        if (threadIdx.x < 2 * OUTD)
            atomAddF(&stats[threadIdx.x], sred[threadIdx.x]);
    }
}

// ---------------------------------------------------------------------------
// stats {sum[C], sumsq[C]} + gamma/beta -> {scale[C], shift[C]}
__global__ void bn_prep(const float* __restrict__ stats,
                        const float* __restrict__ gamma,
                        const float* __restrict__ beta,
                        float* __restrict__ ss, int C) {
    const int c = threadIdx.x;
    if (c < C) {
        const float invN  = 1.0f / (float)NNODES;
        const float mean  = stats[c] * invN;
        const float var   = stats[C + c] * invN - mean * mean;
        const float scale = gamma[c] * rsqrtf(var + 1e-5f);
        ss[c]     = scale;
        ss[C + c] = beta[c] - mean * scale;
    }
}

// H = relu(H * scale[c] + shift[c]) over N x 128, float4-vectorized (b128).
__global__ __launch_bounds__(256) void bn_relu(float* __restrict__ H,
                                               const float* __restrict__ ss) {
    const int i  = blockIdx.x * 256 + threadIdx.x;   // one float4 group
    const int cg = i & 31;                           // 128/4 column groups
    const float4 h  = ((const float4*)H)[i];
    const float4 sc = ((const float4*)ss)[cg];
    const float4 sh = ((const float4*)ss)[32 + cg];
    float4 o;
    o.x = fmaxf(fmaf(h.x, sc.x, sh.x), 0.f);
    o.y = fmaxf(fmaf(h.y, sc.y, sh.y), 0.f);
    o.z = fmaxf(fmaf(h.z, sc.z, sh.z), 0.f);
    o.w = fmaxf(fmaf(h.w, sc.w, sh.w), 0.f);
    ((float4*)H)[i] = o;
}

// ---------------------------------------------------------------------------
extern "C" void kernel_launch(void* const* d_in, const int* in_sizes, int n_in,
                              void* d_out, int out_size, void* d_ws, size_t ws_size,
                              hipStream_t stream) {
    const float*     x    = (const float*)d_in[0];
    const long long* ei   = (const long long*)d_in[1];   // int64 (2, E)
    const long long* rowi = ei;
    const long long* coli = ei + NEDGES;
    const float* W1 = (const float*)d_in[2];
    const float* b1 = (const float*)d_in[3];
    const float* g1 = (const float*)d_in[4];
    const float* e1 = (const float*)d_in[5];
    const float* W2 = (const float*)d_in[6];
    const float* b2 = (const float*)d_in[7];
    const float* g2 = (const float*)d_in[8];
    const float* e2 = (const float*)d_in[9];
    const float* W3 = (const float*)d_in[10];
    const float* b3 = (const float*)d_in[11];

    float* ws    = (float*)d_ws;
    float* rdeg  = ws;                                   // N floats
    float* buf0  = ws + 100352;                          // N*128 (aligned base)
    float* buf1  = buf0 + (size_t)NNODES * 128;          // N*128
    float* stats = buf1 + (size_t)NNODES * 128;          // 256 floats
    float* ss    = stats + 256;                          // 256 floats

    const int SC64_BLOCKS  = NEDGES / 16;     // E*16/256  = 100000
    const int SC128_BLOCKS = NEDGES / 8;      // E*32/256  = 200000
    const int GEMM_BLOCKS  = (MTILES + 7) / 8;// 782 (8 waves/block, 1 mtile/wave)
    const int BN_BLOCKS    = NNODES / 8;      // N*128/(4*256) = 12500

    // degree -> reciprocal (shared by all 3 layers)
    hipMemsetAsync(rdeg, 0, NNODES * sizeof(float), stream);
    degree_kernel<<<NEDGES / 256, 256, 0, stream>>>(rowi, rdeg);
    rdeg_kernel<<<(NNODES + 255) / 256, 256, 0, stream>>>(rdeg, NNODES);

    // ----- layer 1: x(64) -> h1(128) -----
    hipMemsetAsync(buf0, 0, (size_t)NNODES * 64 * sizeof(float), stream);
    scatter_add_kernel<64><<<SC64_BLOCKS, 256, 0, stream>>>(rowi, coli, x, buf0);
    hipMemsetAsync(stats, 0, 256 * sizeof(float), stream);
    gemm_bias_stats<64, 128, true><<<GEMM_BLOCKS, 256, 0, stream>>>(
        buf0, rdeg, W1, b1, buf1, stats);
    bn_prep<<<1, 128, 0, stream>>>(stats, g1, e1, ss, 128);
    bn_relu<<<BN_BLOCKS, 256, 0, stream>>>(buf1, ss);

    // ----- layer 2: h1(128) -> h2(128) -----
    hipMemsetAsync(buf0, 0, (size_t)NNODES * 128 * sizeof(float), stream);
    scatter_add_kernel<128><<<SC128_BLOCKS, 256, 0, stream>>>(rowi, coli, buf1, buf0);
    hipMemsetAsync(stats, 0, 256 * sizeof(float), stream);
    gemm_bias_stats<128, 128, true><<<GEMM_BLOCKS, 256, 0, stream>>>(
        buf0, rdeg, W2, b2, buf1, stats);
    bn_prep<<<1, 128, 0, stream>>>(stats, g2, e2, ss, 128);
    bn_relu<<<BN_BLOCKS, 256, 0, stream>>>(buf1, ss);

    // ----- layer 3: h2(128) -> out(64), no BN -----
    hipMemsetAsync(buf0, 0, (size_t)NNODES * 128 * sizeof(float), stream);
    scatter_add_kernel<128><<<SC128_BLOCKS, 256, 0, stream>>>(rowi, coli, buf1, buf0);
    gemm_bias_stats<128, 64, false><<<GEMM_BLOCKS, 256, 0, stream>>>(
        buf0, rdeg, W3, b3, (float*)d_out, nullptr);
}